// Model_73383811219851
// MI455X (gfx1250) — compile-verified
//
#include <hip/hip_runtime.h>
#include <hip/hip_bf16.h>

// ---------------------------------------------------------------------------
// Adaptive wavelet-AR predictor, fully sequential (w carried across batch &
// steps) -> one persistent wave32 workgroup on one WGP.
//   per iteration:
//     - tail-window a-trous convs (levels 0..2) into LDS
//     - build augmented regression matrix Baug (60 x 16), col14 = s
//     - Gaug = Baug^T Baug via chained V_WMMA_F32_16X16X4_F32 (15 K-chunks)
//       (diagonal ridge folded into the D-matrix store)
//     - 14x14 fully-unrolled register Gauss-Jordan, pivot broadcasts via
//       v_readlane_b32 (SGPR scalar operands), no dynamic register indexing
//     - w update (mu = 0.5), pred = dot(B[0], w), append pred to signal
// ---------------------------------------------------------------------------

typedef __attribute__((ext_vector_type(2)))  float    v2f;
typedef __attribute__((ext_vector_type(8)))  float    v8f;

// prefix sums of DB4_LO / DB4_HI (the reference's cumsum quirk); level-j filter
// tap p is  2^{-j/2} * P[p >> j],  length 7*2^j + 1.
__device__ constexpr float PH[8] = {
    -0.010597401784997278f,  0.022285609881985667f,  0.05312699171797263f,
    -0.13390782000090851f,  -0.16189158941789236f,   0.46898917851169805f,
     1.1838357490642396f,    1.4142135623730948f };
__device__ constexpr float PG[8] = {
    -0.23037781330885523f,   0.48446875724368627f,  -0.14641201068590413f,
    -0.17439578010288798f,   0.012639031615993157f,  0.04348041345198012f,
     0.010597401784997177f, -1.0097419586828951e-16f };

__device__ constexpr int OFFc[14] = {0,1,2,3, 0,2,4,6, 0,4,8,12, 0,8};
__device__ constexpr int LEVc[14] = {0,0,0,0, 1,1,1,1, 2,2,2,2, 3,3};

#define NF1 0.70710678118654752f
#define NF2 0.5f

// broadcast lane l's value (literal l) -> v_readlane_b32 into SGPR
__device__ __forceinline__ float rdlane(float v, int l) {
  union { float f; int i; } u;
  u.f = v;
  u.i = __builtin_amdgcn_readlane(u.i, l);
  return u.f;
}

__global__ __launch_bounds__(32, 1)
void Model_73383811219851_kernel(const float* __restrict__ x,
                                 float* __restrict__ out) {
  // signal tail buffer: 1024 samples + up to 16 appended predictions
  alignas(16) __shared__ float buf[1040];
  __shared__ float C1[120];          // c1 over t in [928,1047]
  __shared__ float C2[100];          // c2 over t in [938,1037]
  __shared__ float W0a[72];          // W0 over t in [952,1023]
  __shared__ float W1a[72];
  __shared__ float W2a[72];
  __shared__ float C3a[72];
  __shared__ float Baug[60][16];     // augmented regression matrix, col14 = s
  __shared__ float Gaug[16][16];     // Baug^T Baug (G | rhs)

  const int lane = threadIdx.x;
  float wreg = 0.0f;                 // lane i < 14 owns w[i], carried forever
  bool  first = true;                // init==0 only before the very first fit

  for (int b = 0; b < 256; ++b) {
    const float* xb = x + (size_t)b * 1024;
    for (int i = lane; i < 256; i += 32)
      ((float4*)buf)[i] = ((const float4*)xb)[i];
    if (b + 1 < 256) __builtin_prefetch(x + (size_t)(b + 1) * 1024, 0, 1);
    __syncthreads();

    int base = 0;
    for (int step = 0; step < 16; ++step) {
      // ---- stage A: level-0 convs straight from sig ----------------------
      // sig(t) = buf[base+t] for t<=1023, else 0 ('same'-mode zero pad)
      for (int i = lane; i < 120; i += 32) {
        int t = 928 + i;
        float ch = 0.0f, cw = 0.0f;
#pragma unroll
        for (int k = 0; k < 8; ++k) {
          int u = t + 3 - k;
          float sv = (u <= 1023) ? buf[base + u] : 0.0f;
          ch += PH[k] * sv;
          cw += PG[k] * sv;
        }
        C1[i] = ch;
        if (t >= 952 && t <= 1023) W0a[t - 952] = cw;
      }
      __syncthreads();

      // ---- stage B: level-1 (15 taps, half = 7) --------------------------
      for (int i = lane; i < 100; i += 32) {
        int t = 938 + i;
        float ch = 0.0f, cw = 0.0f;
#pragma unroll
        for (int k = 0; k < 15; ++k) {
          float cv = C1[t + 7 - k - 928];
          ch += (NF1 * PH[k >> 1]) * cv;
          cw += (NF1 * PG[k >> 1]) * cv;
        }
        C2[i] = ch;
        if (t >= 952 && t <= 1023) W1a[t - 952] = cw;
      }
      __syncthreads();

      // ---- stage C: level-2 (29 taps, half = 14) -------------------------
      for (int i = lane; i < 72; i += 32) {
        int t = 952 + i;
        float ch = 0.0f, cw = 0.0f;
#pragma unroll
        for (int k = 0; k < 29; ++k) {
          float cv = C2[t + 14 - k - 938];
          ch += (NF2 * PH[k >> 2]) * cv;
          cw += (NF2 * PG[k >> 2]) * cv;
        }
        C3a[i] = ch;
        W2a[i] = cw;
      }
      __syncthreads();

      // ---- build Baug (60 x 16): cols 0..13 coeffs, col14 = s, col15 = 0 -
      for (int r = lane; r < 60; r += 32) {
        int t  = 1023 - r;
        int ti = t - 952;
#pragma unroll
        for (int c = 0; c < 14; ++c) {
          int idx = ti - OFFc[c];
          float v = (LEVc[c] == 0) ? W0a[idx]
                  : (LEVc[c] == 1) ? W1a[idx]
                  : (LEVc[c] == 2) ? W2a[idx]
                  :                  C3a[idx];
          Baug[r][c] = v;
        }
        Baug[r][14] = buf[base + t];   // s[r] = sig[1023 - r]
        Baug[r][15] = 0.0f;
      }
      __syncthreads();

      // ---- Gaug = Baug^T Baug via WMMA -----------------------------------
      v8f acc = {0.f,0.f,0.f,0.f,0.f,0.f,0.f,0.f};
      const int n = lane & 15;
      {
        // A(16x4) = chunk^T and B(4x16) = chunk share the same per-lane data
        // (Gram product is symmetric): lane holds chunk[k0+koff+{0,1}][lane&15]
        const int koff = (lane < 16) ? 0 : 2;
#pragma unroll
        for (int kb = 0; kb < 15; ++kb) {
          v2f fr;
          fr.x = Baug[kb * 4 + koff + 0][n];
          fr.y = Baug[kb * 4 + koff + 1][n];
          acc = __builtin_amdgcn_wmma_f32_16x16x4_f32(
              false, fr, false, fr, (short)0, acc, false, false);
        }
      }
      // D layout: lanes 0-15 -> rows 0..7 col=lane; lanes 16-31 -> rows 8..15.
      // Fold the tiny ridge (~rcond cutoff) into the diagonal on store.
      {
        const int mb = (lane < 16) ? 0 : 8;
#pragma unroll
        for (int r = 0; r < 8; ++r) {
          float v = acc[r];
          if (n == mb + r) v += 1e-7f * fabsf(v);   // constant-index diag test
          Gaug[mb + r][n] = v;
        }
      }
      __syncthreads();

      // ---- 14x14 register Gauss-Jordan, fully unrolled; pivot broadcasts
      //      via v_readlane_b32; no lane-varying register indexing ----------
      const int ii = (lane < 14) ? lane : 13;   // spare lanes shadow row 13
      float row[15];
#pragma unroll
      for (int j = 0; j < 15; ++j) row[j] = Gaug[ii][j];

      float mydiag = 1.0f;
#pragma unroll
      for (int k = 0; k < 14; ++k) {
        // lane k's diagonal row[k] is final entering step k: capture it with a
        // constant register index (single cndmask).
        mydiag = (ii == k) ? row[k] : mydiag;
        float pk = rdlane(row[k], k);
        float f  = (ii == k) ? 0.0f : (row[k] / pk);
        // pivot-row entries at j < k are exactly zero -> triangular update
#pragma unroll
        for (int j = k; j < 15; ++j)
          row[j] -= f * rdlane(row[j], k);
      }
      float wnew = row[14] / mydiag;

      // ---- weight update (mu = 0.5) + prediction -------------------------
      wreg = first ? wnew : (0.5f * wreg + 0.5f * wnew);

      float p = (lane < 14) ? (Baug[0][lane] * wreg) : 0.0f;
#pragma unroll
      for (int o = 16; o > 0; o >>= 1) p += __shfl_xor(p, o);

      if (lane == 0) {
        buf[base + 1024] = p;                       // append pred (shift)
        out[b * 16 + step] = p;
      }
      __syncthreads();
      base += 1;
      first = false;
    }
    __syncthreads();
  }
}

extern "C" void kernel_launch(void* const* d_in, const int* in_sizes, int n_in,
                              void* d_out, int out_size, void* d_ws, size_t ws_size,
                              hipStream_t stream) {
  (void)in_sizes; (void)n_in; (void)d_ws; (void)ws_size; (void)out_size;
  const float* x = (const float*)d_in[0];
  float* out = (float*)d_out;
  // entirely sequential recurrence -> one persistent wave32 workgroup
  hipLaunchKernelGGL(Model_73383811219851_kernel, dim3(1), dim3(32), 0, stream,
                     x, out);
}